// SpatialSparsity_7413113552936
// MI455X (gfx1250) — compile-verified
//
#include <hip/hip_runtime.h>
#include <cstdint>

// Spatial winner-take-all (k=1) for (64, 384, 48, 48) fp32.
// Per row of 2304 floats: rowmax, mask-out everything below it.
// Memory-bound: ~453 MB total traffic -> ~19.5 us at 23.3 TB/s.
// CDNA5 path: async global->LDS b128 streaming (ASYNCcnt) + wave32 shfl
// reduce + non-temporal b128 stores (output is write-once, keep it out of L2).

#define AS1 __attribute__((address_space(1)))
#define AS3 __attribute__((address_space(3)))

typedef int   v4i __attribute__((ext_vector_type(4)));
typedef float v4f __attribute__((ext_vector_type(4)));

#define ROW   2304   // 48*48 elements per (b,c) row
#define HALF  1152
#define TPB   288    // 9 wave32s; 288 lanes * 8 floats = 2304
#define NWAVE (TPB / 32)

// Generic (flat) pointer to an LDS object -> 32-bit LDS address.
// Per CDNA5 ISA aperture rules, addr[31:0] of a flat LDS pointer is the LDS
// byte offset, so truncation is the canonical cast.
__device__ __forceinline__ AS3 v4i* to_lds(const void* p) {
  return (AS3 v4i*)(uint32_t)(uint64_t)p;
}

// Generic pointer to global memory -> addrspace(1) v4i pointer (drops const).
__device__ __forceinline__ AS1 v4i* to_global(const void* p) {
  return (AS1 v4i*)(uint64_t)p;
}

__global__ __launch_bounds__(TPB) void spatial_wta_kernel(
    const float* __restrict__ act,    // (rows, 2304)
    float* __restrict__ res,          // (rows, 2304)
    float* __restrict__ winners) {    // (rows,)
  __shared__ __align__(16) float srow[ROW];
  __shared__ float swav[NWAVE];

  const int row = blockIdx.x;
  const int tid = threadIdx.x;
  const float* g = act + (size_t)row * ROW;

  // ---- Stage the whole row into LDS with async b128 copies (2 per lane) ----
  {
    const float* g0 = g + tid * 4;
    const float* g1 = g + HALF + tid * 4;
    AS3 v4i* l0 = to_lds(&srow[tid * 4]);
    AS3 v4i* l1 = to_lds(&srow[HALF + tid * 4]);
#if __has_builtin(__builtin_amdgcn_global_load_async_to_lds_b128)
    __builtin_amdgcn_global_load_async_to_lds_b128(to_global(g0), l0, 0, 0);
    __builtin_amdgcn_global_load_async_to_lds_b128(to_global(g1), l1, 0, 0);
#else
    asm volatile("global_load_async_to_lds_b128 %0, %1, off"
                 :: "v"((uint32_t)(uint64_t)(void*)l0), "v"(g0) : "memory");
    asm volatile("global_load_async_to_lds_b128 %0, %1, off"
                 :: "v"((uint32_t)(uint64_t)(void*)l1), "v"(g1) : "memory");
#endif
#if __has_builtin(__builtin_amdgcn_s_wait_asynccnt)
    __builtin_amdgcn_s_wait_asynccnt(0);
#else
    asm volatile("s_wait_asynccnt 0" ::: "memory");
#endif
  }
  // Each wave waited for its own async loads; barrier covers the other waves.
  __syncthreads();

  // ---- Per-thread max over its 8 staged elements (2x ds_load_b128) ----
  v4f a0 = *(const AS3 v4f*)to_lds(&srow[tid * 4]);
  v4f a1 = *(const AS3 v4f*)to_lds(&srow[HALF + tid * 4]);
  float m = fmaxf(fmaxf(fmaxf(a0.x, a0.y), fmaxf(a0.z, a0.w)),
                  fmaxf(fmaxf(a1.x, a1.y), fmaxf(a1.z, a1.w)));

  // ---- wave32 butterfly max ----
  #pragma unroll
  for (int off = 16; off > 0; off >>= 1)
    m = fmaxf(m, __shfl_xor(m, off, 32));

  if ((tid & 31) == 0) swav[tid >> 5] = m;
  __syncthreads();

  if (tid == 0) {
    float r = swav[0];
    #pragma unroll
    for (int w = 1; w < NWAVE; ++w) r = fmaxf(r, swav[w]);
    swav[0] = r;
    __builtin_nontemporal_store(r, winners + row);   // per-row winner value
  }
  __syncthreads();
  const float rowmax = swav[0];

  // ---- Masked write-back: keep all positions equal to rowmax (tie semantics
  //      identical to jnp.where(y < winners, 0, 1) * y) ----
  a0.x = (a0.x < rowmax) ? 0.0f : a0.x;
  a0.y = (a0.y < rowmax) ? 0.0f : a0.y;
  a0.z = (a0.z < rowmax) ? 0.0f : a0.z;
  a0.w = (a0.w < rowmax) ? 0.0f : a0.w;
  a1.x = (a1.x < rowmax) ? 0.0f : a1.x;
  a1.y = (a1.y < rowmax) ? 0.0f : a1.y;
  a1.z = (a1.z < rowmax) ? 0.0f : a1.z;
  a1.w = (a1.w < rowmax) ? 0.0f : a1.w;

  // Write-once stream: non-temporal b128 stores keep 226 MB of output from
  // evicting the inbound activation stream out of L2.
  float* o = res + (size_t)row * ROW;
  __builtin_nontemporal_store(a0, (v4f*)(o + tid * 4));
  __builtin_nontemporal_store(a1, (v4f*)(o + HALF + tid * 4));
}

extern "C" void kernel_launch(void* const* d_in, const int* in_sizes, int n_in,
                              void* d_out, int out_size, void* d_ws, size_t ws_size,
                              hipStream_t stream) {
  const float* act = (const float*)d_in[0];
  float* out = (float*)d_out;

  const int n    = in_sizes[0];   // 64*384*48*48
  const int rows = n / ROW;       // 64*384 = 24576

  float* res     = out;           // first output, n elements
  float* winners = out + n;       // second output, rows elements

  spatial_wta_kernel<<<rows, TPB, 0, stream>>>(act, res, winners);
}